// VectorQuantizer_86998857548092
// MI455X (gfx1250) — compile-verified
//
#include <hip/hip_runtime.h>
#include <hip/hip_bf16.h>
#include <math.h>

typedef __attribute__((ext_vector_type(2))) float v2f;
typedef __attribute__((ext_vector_type(8))) float v8f;

#define T_TOK 16384
#define K_CB  4096
#define D_EMB 256

// ---------------- kernel 1: codebook row norms ----------------
__global__ void vq_cnorm(const float* __restrict__ cb, float* __restrict__ cnorm) {
    int k = blockIdx.x * blockDim.x + threadIdx.x;
    if (k < K_CB) {
        const float4* row = (const float4*)(cb + (size_t)k * D_EMB);
        float s = 0.f;
#pragma unroll 8
        for (int i = 0; i < D_EMB / 4; ++i) {
            float4 v = row[i];
            s += v.x * v.x + v.y * v.y + v.z * v.z + v.w * v.w;
        }
        cnorm[k] = s;
    }
}

// ---------------- kernel 2: zero encodings + counts ----------------
__global__ void vq_fill(float4* __restrict__ enc4, int* __restrict__ counts) {
    size_t i = (size_t)blockIdx.x * blockDim.x + threadIdx.x;
    size_t stride = (size_t)gridDim.x * blockDim.x;
    const size_t n4 = (size_t)T_TOK * K_CB / 4;
    float4 z4 = make_float4(0.f, 0.f, 0.f, 0.f);
    for (size_t p = i; p < n4; p += stride) enc4[p] = z4;
    if (i < K_CB) counts[i] = 0;
}

// ---------------- kernel 3: WMMA distance GEMM + argmin ----------------
// block = 128 threads (4 waves), handles 64 tokens; wave w -> tokens [64*blk + 16w, +16)
__global__ void __launch_bounds__(128)
vq_argmin(const float* __restrict__ z, const float* __restrict__ cb,
          const float* __restrict__ cnorm, int* __restrict__ idxOut) {
    __shared__ float zs[64 * D_EMB];   // 64 KB z tile

    // cooperative staged load of z tile (coalesced float4)
    {
        const float4* z4 = (const float4*)(z + (size_t)blockIdx.x * 64 * D_EMB);
        float4* zs4 = (float4*)zs;
#pragma unroll
        for (int i = threadIdx.x; i < 64 * D_EMB / 4; i += 128) zs4[i] = z4[i];
    }
    __syncthreads();

    const int lane = threadIdx.x & 31;
    const int wv   = threadIdx.x >> 5;
    const int lmod = lane & 15;        // N column (B/C) or M row (A)
    const int lhi  = lane >> 4;        // K-half selector / M-half selector
    const int klo  = lhi * 2;          // A/B: k offset within 4-wide chunk

    // A base for this wave's M-tile: row = lmod, element klo within each d-chunk
    const float* zsA = &zs[((wv * 16 + lmod) * D_EMB) + klo];

    float best[8];
    int   bidx[8];
#pragma unroll
    for (int r = 0; r < 8; ++r) { best[r] = 3.4e38f; bidx[r] = 0; }

    for (int ng = 0; ng < K_CB / 64; ++ng) {    // 64 columns per group (4 N-tiles)
        const int n0 = ng * 64;
        v8f acc[4];
#pragma unroll
        for (int j = 0; j < 4; ++j) acc[j] = (v8f){};

        const float* bp[4];
#pragma unroll
        for (int j = 0; j < 4; ++j)
            bp[j] = cb + (size_t)(n0 + 16 * j + lmod) * D_EMB + klo;

#pragma unroll 4
        for (int d = 0; d < D_EMB; d += 4) {
            v2f a = *(const v2f*)(zsA + d);
#pragma unroll
            for (int j = 0; j < 4; ++j) {
                v2f b = *(const v2f*)(bp[j] + d);
                acc[j] = __builtin_amdgcn_wmma_f32_16x16x4_f32(
                    false, a, false, b, (short)0, acc[j], false, false);
            }
        }

        // dist = cnorm[n] - 2*dot ; C layout: lane holds N=lmod, rows r + 8*lhi
#pragma unroll
        for (int j = 0; j < 4; ++j) {
            const int n = n0 + 16 * j + lmod;
            const float cn = cnorm[n];
#pragma unroll
            for (int r = 0; r < 8; ++r) {
                float dist = cn - 2.0f * acc[j][r];
                if (dist < best[r]) { best[r] = dist; bidx[r] = n; }
            }
        }
    }

    // reduce across the 16-lane half (xor masks 1,2,4,8 keep bit4 fixed)
#pragma unroll
    for (int off = 1; off < 16; off <<= 1) {
#pragma unroll
        for (int r = 0; r < 8; ++r) {
            float ov = __shfl_xor(best[r], off, 32);
            int   oi = __shfl_xor(bidx[r], off, 32);
            if (ov < best[r] || (ov == best[r] && oi < bidx[r])) {
                best[r] = ov; bidx[r] = oi;
            }
        }
    }

    if (lmod == 0) {
        const int mbase = blockIdx.x * 64 + wv * 16 + lhi * 8;
#pragma unroll
        for (int r = 0; r < 8; ++r) idxOut[mbase + r] = bidx[r];
    }
}

// ---------------- kernel 4: gather + outputs ----------------
__global__ void __launch_bounds__(256)
vq_scatter(const float* __restrict__ z, const float* __restrict__ cb,
           const int* __restrict__ idx, float* __restrict__ outQ,
           float* __restrict__ outEnc, float* __restrict__ outIdx,
           float* __restrict__ psum, int* __restrict__ counts) {
    const int t = blockIdx.x;
    const int d = threadIdx.x;
    const int k = idx[t];
    const float q  = cb[(size_t)k * D_EMB + d];
    const float zz = z[(size_t)t * D_EMB + d];
    outQ[(size_t)t * D_EMB + d] = q;
    float v = (q - zz) * (q - zz);
#pragma unroll
    for (int off = 16; off > 0; off >>= 1) v += __shfl_down(v, off, 32);
    __shared__ float red[8];
    const int lane = threadIdx.x & 31, wv = threadIdx.x >> 5;
    if (lane == 0) red[wv] = v;
    __syncthreads();
    if (threadIdx.x == 0) {
        float s = 0.f;
#pragma unroll
        for (int i = 0; i < 8; ++i) s += red[i];
        psum[t] = s;
        atomicAdd(&counts[k], 1);
        outEnc[(size_t)t * K_CB + k] = 1.0f;
        outIdx[t] = (float)k;
    }
}

// ---------------- kernel 5: loss + perplexity ----------------
__global__ void __launch_bounds__(256)
vq_finalize(const int* __restrict__ counts, const float* __restrict__ psum,
            float* __restrict__ outLoss, float* __restrict__ outPpl) {
    __shared__ float sh[256];
    float h = 0.f;
    for (int k = threadIdx.x; k < K_CB; k += 256) {
        float p = (float)counts[k] * (1.0f / T_TOK);
        h += p * logf(p + 1e-10f);
    }
    sh[threadIdx.x] = h;
    __syncthreads();
    for (int s = 128; s > 0; s >>= 1) {
        if (threadIdx.x < s) sh[threadIdx.x] += sh[threadIdx.x + s];
        __syncthreads();
    }
    const float ppl = expf(-sh[0]);
    __syncthreads();
    float s2 = 0.f;
    for (int t = threadIdx.x; t < T_TOK; t += 256) s2 += psum[t];
    sh[threadIdx.x] = s2;
    __syncthreads();
    for (int s = 128; s > 0; s >>= 1) {
        if (threadIdx.x < s) sh[threadIdx.x] += sh[threadIdx.x + s];
        __syncthreads();
    }
    if (threadIdx.x == 0) {
        const float mse = sh[0] / ((float)T_TOK * (float)D_EMB);
        *outLoss = 1.25f * mse;   // q_latent + 0.25 * e_latent (equal forward values)
        *outPpl  = ppl;
    }
}

extern "C" void kernel_launch(void* const* d_in, const int* in_sizes, int n_in,
                              void* d_out, int out_size, void* d_ws, size_t ws_size,
                              hipStream_t stream) {
    const float* z  = (const float*)d_in[0];   // [16,1024,256] f32
    const float* cb = (const float*)d_in[1];   // [4096,256] f32
    float* out = (float*)d_out;

    // output tuple layout (flat, return order)
    float* outLoss = out;                                   // 1
    float* outQ    = out + 1;                               // 16*1024*256
    float* outPpl  = out + 1 + (size_t)T_TOK * D_EMB;       // 1
    float* outEnc  = outPpl + 1;                            // T*K
    float* outIdx  = outEnc + (size_t)T_TOK * K_CB;         // T

    // workspace layout
    float* cnorm  = (float*)d_ws;                           // 4096 f32
    int*   idxBuf = (int*)(cnorm + K_CB);                   // 16384 i32
    int*   counts = (int*)(idxBuf + T_TOK);                 // 4096 i32
    float* psum   = (float*)(counts + K_CB);                // 16384 f32

    vq_cnorm<<<K_CB / 256, 256, 0, stream>>>(cb, cnorm);
    vq_fill<<<2048, 256, 0, stream>>>((float4*)outEnc, counts);
    vq_argmin<<<T_TOK / 64, 128, 0, stream>>>(z, cb, cnorm, idxBuf);
    vq_scatter<<<T_TOK, 256, 0, stream>>>(z, cb, idxBuf, outQ, outEnc, outIdx,
                                          psum, counts);
    vq_finalize<<<1, 256, 0, stream>>>(counts, psum, outLoss, outPpl);
}